// DistanceMatrixLayer_41875931136253
// MI455X (gfx1250) — compile-verified
//
#include <hip/hip_runtime.h>

// ---------------------------------------------------------------------------
// Pairwise squared Euclidean distance via ||a||^2 + ||b||^2 - 2 a.b,
// with the a.b GEMM done on CDNA5 matrix units in exact fp32
// (V_WMMA_F32_16X16X4_F32). B=4, N=M=1024, D=64, fp32 everywhere.
// ---------------------------------------------------------------------------

typedef __attribute__((ext_vector_type(2))) float v2f;
typedef __attribute__((ext_vector_type(8))) float v8f;

#define BATCH 4
#define NN 1024
#define MM 1024
#define DD 64

// Cross-lane gather via LDS-permute hardware (no LDS memory touched).
__device__ __forceinline__ float lane_gather(float v, int srcLane) {
  return __int_as_float(
      __builtin_amdgcn_ds_bpermute(srcLane << 2, __float_as_int(v)));
}

__global__ __launch_bounds__(256) void dist2_wmma_f32(
    const float* __restrict__ t1,   // [B, N, D]
    const float* __restrict__ t2,   // [B, M, D]
    float* __restrict__ out) {      // [B, N, M]
  const int lane = threadIdx.x & 31;
  const int wave = threadIdx.x >> 5;

  // One wave-task = one 16 (rows) x 64 (cols) output strip of one batch.
  // tasks = 4 batches * 64 i-tiles * 16 j-groups = 4096 = 512 blocks * 8 waves.
  const int task = blockIdx.x * 8 + wave;
  const int jg   = task & 15;         // j-group (64 cols each)
  const int it   = (task >> 4) & 63;  // i-tile  (16 rows each)
  const int b    = task >> 10;        // batch

  const int h = lane >> 4;   // which half of the wave (selects k-pair)
  const int m = lane & 15;   // A-row / B-col within tile

  const int i0 = it * 16;
  const int j0 = jg * 64;

  // Fragment rows: A lane holds row (i0+m), k = kb + 2h + {0,1}
  //                B lane holds col (j0+t*16+m), same k pair (B = traj2^T).
  const float* arow  = t1 + (size_t)(b * NN + i0 + m) * DD;
  const float* brow0 = t2 + (size_t)(b * MM + j0 + m) * DD;
  const float* brow1 = brow0 + 16 * DD;
  const float* brow2 = brow0 + 32 * DD;
  const float* brow3 = brow0 + 48 * DD;

  v8f acc0 = {}, acc1 = {}, acc2 = {}, acc3 = {};
  float na = 0.f, nb0 = 0.f, nb1 = 0.f, nb2 = 0.f, nb3 = 0.f;

#pragma unroll
  for (int kb = 0; kb < DD; kb += 4) {
    const int k = kb + 2 * h;
    v2f a  = *(const v2f*)(arow  + k);
    v2f q0 = *(const v2f*)(brow0 + k);
    v2f q1 = *(const v2f*)(brow1 + k);
    v2f q2 = *(const v2f*)(brow2 + k);
    v2f q3 = *(const v2f*)(brow3 + k);

    // Partial squared norms (each half-wave covers its half of k).
    na  += a.x * a.x + a.y * a.y;
    nb0 += q0.x * q0.x + q0.y * q0.y;
    nb1 += q1.x * q1.x + q1.y * q1.y;
    nb2 += q2.x * q2.x + q2.y * q2.y;
    nb3 += q3.x * q3.x + q3.y * q3.y;

    // D = A x B + C, exact fp32 on the matrix unit.
    acc0 = __builtin_amdgcn_wmma_f32_16x16x4_f32(false, a, false, q0,
                                                 (short)0, acc0, false, false);
    acc1 = __builtin_amdgcn_wmma_f32_16x16x4_f32(false, a, false, q1,
                                                 (short)0, acc1, false, false);
    acc2 = __builtin_amdgcn_wmma_f32_16x16x4_f32(false, a, false, q2,
                                                 (short)0, acc2, false, false);
    acc3 = __builtin_amdgcn_wmma_f32_16x16x4_f32(false, a, false, q3,
                                                 (short)0, acc3, false, false);
  }

  // Combine the two half-wave k-partials: lane L <- L ^ 16.
  na  += lane_gather(na,  lane ^ 16);
  nb0 += lane_gather(nb0, lane ^ 16);
  nb1 += lane_gather(nb1, lane ^ 16);
  nb2 += lane_gather(nb2, lane ^ 16);
  nb3 += lane_gather(nb3, lane ^ 16);
  // Now lane L holds: ||traj1[i0+(L&15)]||^2 in na,
  //                   ||traj2[j0+t*16+(L&15)]||^2 in nb_t.

  float* obase = out + (size_t)b * NN * MM;

#pragma unroll
  for (int v = 0; v < 8; ++v) {
    // C/D layout: VGPR v holds output row (v + 8h), col (lane&15).
    const int mi = v + 8 * h;
    const float n1 = lane_gather(na, mi);  // norm of output row i0+mi
    float* orow = obase + (size_t)(i0 + mi) * MM + j0 + m;
    orow[0]  = fmaxf(fmaf(-2.f, acc0[v], n1 + nb0), 0.f);
    orow[16] = fmaxf(fmaf(-2.f, acc1[v], n1 + nb1), 0.f);
    orow[32] = fmaxf(fmaf(-2.f, acc2[v], n1 + nb2), 0.f);
    orow[48] = fmaxf(fmaf(-2.f, acc3[v], n1 + nb3), 0.f);
  }
}

extern "C" void kernel_launch(void* const* d_in, const int* in_sizes, int n_in,
                              void* d_out, int out_size, void* d_ws,
                              size_t ws_size, hipStream_t stream) {
  (void)in_sizes; (void)n_in; (void)out_size; (void)d_ws; (void)ws_size;
  const float* traj1 = (const float*)d_in[0];  // [4,1024,64]
  const float* traj2 = (const float*)d_in[1];  // [4,1024,64]
  float* out = (float*)d_out;                  // [4,1024,1024]

  // 4096 wave-tasks, 8 waves (256 threads) per block -> 512 blocks.
  dist2_wmma_f32<<<512, 256, 0, stream>>>(traj1, traj2, out);
}